// GraphSage_31980326486699
// MI455X (gfx1250) — compile-verified
//
#include <hip/hip_runtime.h>

typedef __attribute__((ext_vector_type(2))) float v2f;
typedef __attribute__((ext_vector_type(8))) float v8f;

#define HID 64

// ---------------------------------------------------------------- utilities
__global__ void zero_kernel(float* __restrict__ p, long n) {
  long i = (long)blockIdx.x * blockDim.x + threadIdx.x;
  long stride = (long)gridDim.x * blockDim.x;
  for (; i < n; i += stride) p[i] = 0.0f;
}

__global__ void deg_kernel(const int* __restrict__ dst, float* __restrict__ deg, int nE) {
  int i = blockIdx.x * blockDim.x + threadIdx.x;
  if (i < nE) atomicAdd(&deg[dst[i]], 1.0f);
}

__global__ void invdeg_kernel(float* __restrict__ deg, int n) {
  int i = blockIdx.x * blockDim.x + threadIdx.x;
  if (i < n) { float d = deg[i]; deg[i] = d > 0.0f ? 1.0f / d : 0.0f; }
}

// Mean-aggregation scatter: 16 threads per edge, float4 per thread.
// h rows are 64 floats (16B aligned), reads coalesced, atomics hit L2.
__global__ void scatter_kernel(const float* __restrict__ h, const int* __restrict__ src,
                               const int* __restrict__ dst, float* __restrict__ agg, int nE) {
  int t = blockIdx.x * blockDim.x + threadIdx.x;
  int total = nE * 16;
  if (t >= total) return;
  int e = t >> 4, part = t & 15;
  int s = src[e], d = dst[e];
  const float4 v = *(const float4*)(h + (long)s * HID + part * 4);
  float* o = agg + (long)d * HID + part * 4;
  atomicAdd(o + 0, v.x);
  atomicAdd(o + 1, v.y);
  atomicAdd(o + 2, v.z);
  atomicAdd(o + 3, v.w);
}

// ---------------------------------------------------------------- WMMA GEMM
// One K-sweep accumulating A[row0:row0+16, 0:K] @ B[0:K, 0:64] into acc[4]
// using V_WMMA_F32_16X16X4_F32.  A-fragment layout (ISA 7.12.2, 32-bit A 16x4):
//   lanes 0-15  : M=lane,    VGPR0 = K=k+0, VGPR1 = K=k+1
//   lanes 16-31 : M=lane-16, VGPR0 = K=k+2, VGPR1 = K=k+3
// B-fragment mirrors with N=lane&15.  rowScale (inv_deg) folds the mean.
__device__ __forceinline__ void gemm_pass(const float* __restrict__ A, int lda,
                                          const float* __restrict__ rowScale,
                                          const float* __restrict__ B, int K,
                                          int row0, int lane, v8f acc[4]) {
  const int hr = lane & 15;
  const int kb = (lane >> 4) << 1;  // 0 or 2
  const float* arow = A + (long)(row0 + hr) * lda;
  const float s = rowScale ? rowScale[row0 + hr] : 1.0f;
  for (int k = 0; k < K; k += 4) {
    v2f a;
    a[0] = arow[k + kb] * s;
    a[1] = arow[k + kb + 1] * s;
#pragma unroll
    for (int nt = 0; nt < 4; ++nt) {
      v2f b;
      b[0] = B[(long)(k + kb) * HID + nt * 16 + hr];
      b[1] = B[(long)(k + kb + 1) * HID + nt * 16 + hr];
      acc[nt] = __builtin_amdgcn_wmma_f32_16x16x4_f32(
          /*neg_a=*/false, a, /*neg_b=*/false, b,
          /*c_mod=*/(short)0, acc[nt], /*reuse_a=*/false, /*reuse_b=*/false);
    }
  }
}

// C[M,64] = A1@B1 (+ A2@B2) + bias, optional row-L2-normalize + ReLU.
// One wave per 16-row strip; 8 waves (256 thr) per block; EXEC all-1s per wave.
template <bool HAS_A2, bool DO_NORM_RELU>
__global__ void gemm_kernel(const float* __restrict__ A1, const float* __restrict__ rowScale,
                            const float* __restrict__ B1,
                            const float* __restrict__ A2, const float* __restrict__ B2,
                            const float* __restrict__ bias, float* __restrict__ C,
                            int M, int K1) {
  const int lane = threadIdx.x & 31;
  const int tile = blockIdx.x * (blockDim.x >> 5) + (threadIdx.x >> 5);
  const int numTiles = M >> 4;
  if (tile >= numTiles) return;           // whole-wave guard, EXEC stays all-1s
  const int row0 = tile << 4;
  const int hr = lane & 15;

  v8f acc[4];
#pragma unroll
  for (int nt = 0; nt < 4; ++nt) {
    float bv = bias[nt * 16 + hr];
#pragma unroll
    for (int r = 0; r < 8; ++r) acc[nt][r] = bv;
  }

  gemm_pass(A1, K1, rowScale, B1, K1, row0, lane, acc);
  if (HAS_A2) gemm_pass(A2, HID, nullptr, B2, HID, row0, lane, acc);

  if (DO_NORM_RELU) {
    // D-row m = r + 8*(lane>=16) is striped over the 16 lanes of each half-wave
    // at columns {hr, hr+16, hr+32, hr+48}: reduce squares with xor-shuffles
    // over masks 1,2,4,8 (stays inside the 16-lane group).
#pragma unroll
    for (int r = 0; r < 8; ++r) {
      float ss = 0.0f;
#pragma unroll
      for (int nt = 0; nt < 4; ++nt) ss += acc[nt][r] * acc[nt][r];
      ss += __shfl_xor(ss, 1);
      ss += __shfl_xor(ss, 2);
      ss += __shfl_xor(ss, 4);
      ss += __shfl_xor(ss, 8);
      float sc = 1.0f / fmaxf(sqrtf(ss), 1e-12f);
#pragma unroll
      for (int nt = 0; nt < 4; ++nt) acc[nt][r] = fmaxf(acc[nt][r] * sc, 0.0f);
    }
  }

  const int mBase = row0 + ((lane >> 4) << 3);
#pragma unroll
  for (int r = 0; r < 8; ++r) {
    long m = mBase + r;
#pragma unroll
    for (int nt = 0; nt < 4; ++nt) C[m * HID + nt * 16 + hr] = acc[nt][r];
  }
}

// ---------------------------------------------------------------- launcher
extern "C" void kernel_launch(void* const* d_in, const int* in_sizes, int n_in,
                              void* d_out, int out_size, void* d_ws, size_t ws_size,
                              hipStream_t stream) {
  (void)n_in; (void)out_size; (void)ws_size;
  const float* x  = (const float*)d_in[0];
  const int*   ei = (const int*)d_in[1];
  const float* W1 = (const float*)d_in[2];
  const float* b1 = (const float*)d_in[3];
  const float* Wl = (const float*)d_in[4];
  const float* bl = (const float*)d_in[5];
  const float* Wr = (const float*)d_in[6];
  const float* W2 = (const float*)d_in[7];
  const float* b2 = (const float*)d_in[8];

  const int IN_DIM = 128;
  const int N = in_sizes[0] / IN_DIM;            // 100000
  const int E = in_sizes[1] / 2;                 // 1250000
  const int L = in_sizes[4] / (HID * HID);       // 3

  const int* src = ei;
  const int* dst = ei + E;

  // workspace layout (floats): invdeg | hA | hB | agg  (~77 MB)
  float* invdeg = (float*)d_ws;
  float* hA  = invdeg + N;
  float* hB  = hA + (long)N * HID;
  float* agg = hB + (long)N * HID;

  // degree -> inverse degree
  zero_kernel<<<512, 256, 0, stream>>>(invdeg, N);
  deg_kernel<<<(E + 255) / 256, 256, 0, stream>>>(dst, invdeg, E);
  invdeg_kernel<<<(N + 255) / 256, 256, 0, stream>>>(invdeg, N);

  const int numTiles = N / 16;
  const int gemmBlocks = (numTiles + 7) / 8;

  // h = x @ W1 + b1
  gemm_kernel<false, false><<<gemmBlocks, 256, 0, stream>>>(
      x, nullptr, W1, nullptr, nullptr, b1, hA, N, IN_DIM);

  float* hc = hA;
  float* hn = hB;
  for (int l = 0; l < L; ++l) {
    zero_kernel<<<4096, 256, 0, stream>>>(agg, (long)N * HID);
    int total = E * 16;
    scatter_kernel<<<(total + 255) / 256, 256, 0, stream>>>(hc, src, dst, agg, E);
    gemm_kernel<true, true><<<gemmBlocks, 256, 0, stream>>>(
        agg, invdeg, Wl + (long)l * HID * HID,
        hc, Wr + (long)l * HID * HID,
        bl + (long)l * HID, hn, N, HID);
    float* t = hc; hc = hn; hn = t;
  }

  // out = h @ W2 + b2
  gemm_kernel<false, false><<<gemmBlocks, 256, 0, stream>>>(
      hc, nullptr, W2, nullptr, nullptr, b2, (float*)d_out, N, HID);
}